// MyGRU_GCN_Model_18253611008141
// MI455X (gfx1250) — compile-verified
//
#include <hip/hip_runtime.h>
#include <math.h>

// ---------------------------------------------------------------------------
// MyGRU_GCN_Model on MI455X (gfx1250, wave32)
//
// Dominant cost: m1(16x1024) @ W2^T(80000x1024) -> 327 MB fp32 weight stream,
// ~14us at 23.3 TB/s.  Implemented as a skinny fp32 WMMA GEMM using
// V_WMMA_F32_16X16X4_F32 (exact fp32 matrix math on the matrix pipe).
// The GCN branch is linear up to the final Wo dot, so it collapses exactly
// to scalar per-edge scatters (see kernel_launch comments).
// ---------------------------------------------------------------------------

typedef float v2f __attribute__((ext_vector_type(2)));
typedef float v4f __attribute__((ext_vector_type(4)));
typedef float v8f __attribute__((ext_vector_type(8)));

#define NVERT 5000
#define NEDGE 40000
#define BATCH 16
#define INDIM 128
#define HID   512
#define M1DIM 1024
#define M2DIM 80000   // N * GRU_OUT

// workspace layout (float offsets)
#define WS_DEG   0        // 5000  (becomes dinv in place)
#define WS_DVEC  5000     // 5000  (sum of norm per dst)
#define WS_S     10000    // 80000 (16 x 5000)
#define WS_T     90000    // 80000
#define WS_H     170000   // 8192  (next_hidden 16x512)
#define WS_M1    178192   // 16384 (m1 16x1024)
#define WS_SCAL  194576   // 4     (c1, c2, Cbias)

// ---- init: deg=1.0 (self loop), dvec/s/t = 0 ------------------------------
__global__ void k_init(float* ws) {
    int i = blockIdx.x * blockDim.x + threadIdx.x;
    if (i < WS_H) ws[i] = (i < NVERT) ? 1.0f : 0.0f;
}

// ---- degree accumulation over real edges ----------------------------------
__global__ void k_degree(const int* __restrict__ ei, float* __restrict__ deg) {
    int e = blockIdx.x * blockDim.x + threadIdx.x;
    if (e < NEDGE) atomicAdd(&deg[ei[NEDGE + e]], 1.0f);
}

__global__ void k_rsqrt(float* __restrict__ deg) {
    int i = blockIdx.x * blockDim.x + threadIdx.x;
    if (i < NVERT) deg[i] = rsqrtf(deg[i]);   // deg -> dinv in place
}

// ---- collapse GCN weights against Wo:  u = Wg2 @ wo_res, c1 = Wg1.u, ... --
__global__ void k_scalars(const float* __restrict__ Wg1, const float* __restrict__ bg1,
                          const float* __restrict__ Wg2, const float* __restrict__ bg2,
                          const float* __restrict__ Wo,  const float* __restrict__ bo,
                          float* __restrict__ scal) {
    __shared__ float r1[128], r2[128], r3[128];
    int f = threadIdx.x;                       // 128 threads
    const float* wr = Wo + 16;                 // wo_res (128)
    float u = 0.0f;
    for (int g = 0; g < 128; ++g) u += Wg2[f * 128 + g] * wr[g];
    r1[f] = Wg1[f] * u;
    r2[f] = bg1[f] * u;
    r3[f] = bg2[f] * wr[f];
    __syncthreads();
    if (f == 0) {
        float c1 = 0.f, c2 = 0.f, c3 = 0.f;
        for (int i = 0; i < 128; ++i) { c1 += r1[i]; c2 += r2[i]; c3 += r3[i]; }
        scal[0] = c1; scal[1] = c2; scal[2] = c3 + bo[0];
    }
}

// ---- edge pass 1: s[b,dst] += norm * z[b,src];  dvec[dst] += norm ---------
__global__ void k_edge_s(const int* __restrict__ ei, const float* __restrict__ svp,
                         const float* __restrict__ dinv,
                         float* __restrict__ dvec, float* __restrict__ s) {
    int e = blockIdx.x * blockDim.x + threadIdx.x;
    if (e >= NEDGE + NVERT) return;
    int sn, dn;
    if (e < NEDGE) { sn = ei[e]; dn = ei[NEDGE + e]; }
    else           { sn = dn = e - NEDGE; }     // self loop
    float norm = dinv[sn] * dinv[dn];
    atomicAdd(&dvec[dn], norm);
    #pragma unroll
    for (int b = 0; b < BATCH; ++b) {
        float z = svp[b * (3 * NVERT) + sn * 3 + 2];   // svp[b, sn, 2]
        atomicAdd(&s[b * NVERT + dn], norm * z);
    }
}

// ---- edge pass 2: t[b,dst] += norm * s[b,src] -----------------------------
__global__ void k_edge_t(const int* __restrict__ ei, const float* __restrict__ sbuf,
                         const float* __restrict__ dinv, float* __restrict__ t) {
    int e = blockIdx.x * blockDim.x + threadIdx.x;
    if (e >= NEDGE + NVERT) return;
    int sn, dn;
    if (e < NEDGE) { sn = ei[e]; dn = ei[NEDGE + e]; }
    else           { sn = dn = e - NEDGE; }
    float norm = dinv[sn] * dinv[dn];
    #pragma unroll
    for (int b = 0; b < BATCH; ++b)
        atomicAdd(&t[b * NVERT + dn], norm * sbuf[b * NVERT + sn]);
}

// ---- GRU cell with h=0:  next_h = (1-z)*tanh(ig + r*bhh_g) ----------------
__global__ void k_gru(const float* __restrict__ x, const float* __restrict__ Wih,
                      const float* __restrict__ bih, const float* __restrict__ bhh,
                      float* __restrict__ h, float* __restrict__ out_h) {
    int idx = blockIdx.x * blockDim.x + threadIdx.x;
    if (idx >= BATCH * HID) return;
    int b = idx / HID, j = idx % HID;
    const float* xb = x + b * INDIM;
    float ir = bih[j], iz = bih[HID + j], ig = bih[2 * HID + j];
    const float* wr = Wih + (size_t)j * INDIM;
    const float* wz = Wih + (size_t)(HID + j) * INDIM;
    const float* wg = Wih + (size_t)(2 * HID + j) * INDIM;
    for (int f = 0; f < INDIM; ++f) {
        float xv = xb[f];
        ir += xv * wr[f]; iz += xv * wz[f]; ig += xv * wg[f];
    }
    float r  = 1.0f / (1.0f + __expf(-(ir + bhh[j])));
    float zz = 1.0f / (1.0f + __expf(-(iz + bhh[HID + j])));
    float nn = tanhf(ig + r * bhh[2 * HID + j]);
    float hv = (1.0f - zz) * nn;
    h[idx] = hv;
    out_h[idx] = hv;      // second output of the reference tuple
}

// ---- MLP1: m1 = leaky(h @ W1^T + b1) * invs * g1 + be1 --------------------
__global__ void k_mlp1(const float* __restrict__ h, const float* __restrict__ W1,
                       const float* __restrict__ b1, const float* __restrict__ a1,
                       const float* __restrict__ g1, const float* __restrict__ be1,
                       float* __restrict__ m1) {
    int idx = blockIdx.x * blockDim.x + threadIdx.x;
    if (idx >= BATCH * M1DIM) return;
    int b = idx / M1DIM, i = idx % M1DIM;
    const float* hb = h + b * HID;
    const float* w  = W1 + (size_t)i * HID;
    float y = b1[i];
    for (int f = 0; f < HID; ++f) y += hb[f] * w[f];
    // branchless leaky relu:  max(y,0) + a*min(y,0)
    y = fmaxf(y, 0.0f) + a1[i] * fminf(y, 0.0f);
    const float invs = rsqrtf(1.0f + 1e-5f);
    m1[idx] = y * invs * g1[i] + be1[i];
}

// ---- big WMMA GEMM: (80000x1024) @ (1024x16) + fused epilogue -------------
// One wave per 16-row tile of W2 == one output vertex. fp32 WMMA 16x16x4.
// A lane layout (16x4 f32): lane L: m=L&15, h=L>>4 holds {K=2h, K=2h+1} of
// row m -> one float2 per lane. B (4x16) mirrors it with n=L&15 = batch.
// C/D: acc[r] at lane L = element (M = r + 8*(L>>4), N = L&15).
__global__ __launch_bounds__(256)
void k_gemm2(const float* __restrict__ W2, const float* __restrict__ m1,
             const float* __restrict__ b2, const float* __restrict__ a2,
             const float* __restrict__ g2, const float* __restrict__ be2,
             const float* __restrict__ Wo, const float* __restrict__ scal,
             const float* __restrict__ t,  const float* __restrict__ dvec,
             float* __restrict__ out) {
    const int lane = threadIdx.x & 31;
    const int wave = threadIdx.x >> 5;
    const int tile = blockIdx.x * 8 + wave;       // vertex id, 0..4999
    const int i0   = tile * 16;                   // W2 row base (= vertex*GRU_OUT)
    const int mrow = lane & 15;
    const int hh   = lane >> 4;

    const float* aP = W2 + (size_t)(i0 + mrow) * M1DIM + 2 * hh;
    const float* bP = m1 + (size_t)mrow * M1DIM + 2 * hh;   // column n == lane&15 == batch

    v8f acc = {};
    #pragma unroll 8
    for (int k = 0; k < M1DIM; k += 4) {
        v2f a  = *(const v2f*)(aP + k);
        v2f bb = *(const v2f*)(bP + k);
        acc = __builtin_amdgcn_wmma_f32_16x16x4_f32(
                  /*neg_a=*/false, a, /*neg_b=*/false, bb,
                  /*c_mod=*/(short)0, acc, /*reuse_a=*/false, /*reuse_b=*/false);
    }

    // Epilogue: this lane owns rows i = i0 + 8*hh + r, r = 0..7 (consecutive).
    // Vector-load the 8 coefficients of each epilogue array, branchless leaky.
    const int ebase = i0 + 8 * hh;
    v4f b2lo = *(const v4f*)(b2 + ebase),  b2hi = *(const v4f*)(b2 + ebase + 4);
    v4f a2lo = *(const v4f*)(a2 + ebase),  a2hi = *(const v4f*)(a2 + ebase + 4);
    v4f g2lo = *(const v4f*)(g2 + ebase),  g2hi = *(const v4f*)(g2 + ebase + 4);
    v4f e2lo = *(const v4f*)(be2 + ebase), e2hi = *(const v4f*)(be2 + ebase + 4);
    v4f wolo = *(const v4f*)(Wo + 8 * hh), wohi = *(const v4f*)(Wo + 8 * hh + 4);

    const float invs = rsqrtf(1.0f + 1e-5f);
    float part = 0.0f;
    #pragma unroll
    for (int r = 0; r < 8; ++r) {
        float bb = (r < 4) ? b2lo[r & 3] : b2hi[r & 3];
        float aa = (r < 4) ? a2lo[r & 3] : a2hi[r & 3];
        float gg = (r < 4) ? g2lo[r & 3] : g2hi[r & 3];
        float ee = (r < 4) ? e2lo[r & 3] : e2hi[r & 3];
        float ww = (r < 4) ? wolo[r & 3] : wohi[r & 3];
        float v = acc[r] + bb;
        v = fmaxf(v, 0.0f) + aa * fminf(v, 0.0f);   // branchless leaky relu
        v = v * invs * gg + ee;
        part += v * ww;                              // wo_gru[k], k = 8*hh + r
    }
    part += __shfl_xor(part, 16, 32);                // combine the two K-halves

    if (hh == 0) {
        int b = mrow;                                // batch
        float y = part + scal[0] * t[b * NVERT + tile]
                       + scal[1] * dvec[tile]
                       + scal[2];
        out[b * NVERT + tile] = y;
    }
}

// ---------------------------------------------------------------------------
extern "C" void kernel_launch(void* const* d_in, const int* in_sizes, int n_in,
                              void* d_out, int out_size, void* d_ws, size_t ws_size,
                              hipStream_t stream) {
    const float* x   = (const float*)d_in[0];
    const float* svp = (const float*)d_in[1];
    const int*   ei  = (const int*)  d_in[2];
    const float* Wih = (const float*)d_in[3];
    // d_in[4] = Whh: unused (hidden state is zero -> gh = bhh)
    const float* bih = (const float*)d_in[5];
    const float* bhh = (const float*)d_in[6];
    const float* Wg1 = (const float*)d_in[7];
    const float* bg1 = (const float*)d_in[8];
    const float* Wg2 = (const float*)d_in[9];
    const float* bg2 = (const float*)d_in[10];
    const float* W1  = (const float*)d_in[11];
    const float* b1  = (const float*)d_in[12];
    const float* a1  = (const float*)d_in[13];
    const float* g1  = (const float*)d_in[14];
    const float* be1 = (const float*)d_in[15];
    const float* W2  = (const float*)d_in[16];
    const float* b2  = (const float*)d_in[17];
    const float* a2  = (const float*)d_in[18];
    const float* g2  = (const float*)d_in[19];
    const float* be2 = (const float*)d_in[20];
    const float* Wo  = (const float*)d_in[21];
    const float* bo  = (const float*)d_in[22];

    float* out = (float*)d_out;       // [ y (16*5000) | next_hidden (16*512) ]
    float* ws  = (float*)d_ws;

    float* deg  = ws + WS_DEG;
    float* dvec = ws + WS_DVEC;
    float* sbuf = ws + WS_S;
    float* tbuf = ws + WS_T;
    float* hbuf = ws + WS_H;
    float* m1   = ws + WS_M1;
    float* scal = ws + WS_SCAL;

    k_init   <<<(WS_H + 255) / 256, 256, 0, stream>>>(ws);
    k_degree <<<(NEDGE + 255) / 256, 256, 0, stream>>>(ei, deg);
    k_rsqrt  <<<(NVERT + 255) / 256, 256, 0, stream>>>(deg);
    k_scalars<<<1, 128, 0, stream>>>(Wg1, bg1, Wg2, bg2, Wo, bo, scal);
    k_edge_s <<<(NEDGE + NVERT + 255) / 256, 256, 0, stream>>>(ei, svp, deg, dvec, sbuf);
    k_edge_t <<<(NEDGE + NVERT + 255) / 256, 256, 0, stream>>>(ei, sbuf, deg, tbuf);
    k_gru    <<<(BATCH * HID + 255) / 256, 256, 0, stream>>>(x, Wih, bih, bhh, hbuf, out + M2DIM);
    k_mlp1   <<<(BATCH * M1DIM + 255) / 256, 256, 0, stream>>>(hbuf, W1, b1, a1, g1, be1, m1);
    k_gemm2  <<<NVERT / 8, 256, 0, stream>>>(W2, m1, b2, a2, g2, be2, Wo, scal, tbuf, dvec, out);
}